// PNERF_net_87686052315578
// MI455X (gfx1250) — compile-verified
//
#include <hip/hip_runtime.h>

// ---------------- problem constants ----------------
#define N_ATOMS 3000000
#define FN      3000
#define LFRAG   (N_ATOMS / FN)          // 1000 atoms per fragment
#define SUB     100                      // atoms per sub-chain (exact re-association)
#define NSEG    (N_ATOMS / SUB)          // 30000 sub-chains
#define SCAN_T  1024
#define CHUNK   ((NSEG + SCAN_T - 1) / SCAN_T)   // 30 segments per scan thread

typedef __attribute__((ext_vector_type(2))) float v2f;
typedef __attribute__((ext_vector_type(8))) float v8f;

// ---------------- small vector helpers ----------------
struct F3 { float x, y, z; };
__device__ __forceinline__ F3 mkf3(float x, float y, float z) { F3 r; r.x = x; r.y = y; r.z = z; return r; }
__device__ __forceinline__ F3 sub3(F3 a, F3 b) { return mkf3(a.x - b.x, a.y - b.y, a.z - b.z); }
__device__ __forceinline__ F3 add3(F3 a, F3 b) { return mkf3(a.x + b.x, a.y + b.y, a.z + b.z); }
__device__ __forceinline__ F3 scl3(F3 a, float s) { return mkf3(a.x * s, a.y * s, a.z * s); }
__device__ __forceinline__ F3 cross3(F3 a, F3 b) {
    return mkf3(a.y * b.z - a.z * b.y, a.z * b.x - a.x * b.z, a.x * b.y - a.y * b.x);
}
__device__ __forceinline__ F3 unit3(F3 a) {
    float s = rsqrtf(a.x * a.x + a.y * a.y + a.z * a.z);
    return scl3(a, s);
}

// frame_rows(A,B,C): rows = [bc, cross(n,bc), n]
__device__ __forceinline__ void frame_rows(F3 A, F3 B, F3 C, F3& r0, F3& r1, F3& r2) {
    F3 bc = unit3(sub3(C, B));
    F3 n  = unit3(cross3(sub3(B, A), bc));
    r0 = bc;
    r1 = cross3(n, bc);
    r2 = n;
}

// affine element: x -> x @ G + d   (row-vector convention, G row-major)
struct Aff { float G[9]; float d[3]; };

__device__ __forceinline__ Aff aff_identity() {
    Aff a;
    a.G[0] = 1.f; a.G[1] = 0.f; a.G[2] = 0.f;
    a.G[3] = 0.f; a.G[4] = 1.f; a.G[5] = 0.f;
    a.G[6] = 0.f; a.G[7] = 0.f; a.G[8] = 1.f;
    a.d[0] = a.d[1] = a.d[2] = 0.f;
    return a;
}
// apply L first, then R:  G = R.G @ L.G ;  d = R.d @ L.G + L.d
__device__ __forceinline__ Aff aff_combine(const Aff& L, const Aff& R) {
    Aff o;
#pragma unroll
    for (int i = 0; i < 3; ++i)
#pragma unroll
        for (int j = 0; j < 3; ++j)
            o.G[i * 3 + j] = R.G[i * 3 + 0] * L.G[0 + j] +
                             R.G[i * 3 + 1] * L.G[3 + j] +
                             R.G[i * 3 + 2] * L.G[6 + j];
#pragma unroll
    for (int j = 0; j < 3; ++j)
        o.d[j] = R.d[0] * L.G[0 + j] + R.d[1] * L.G[3 + j] + R.d[2] * L.G[6 + j] + L.d[j];
    return o;
}
__device__ __forceinline__ void aff_load(Aff& a, const float* p) {
#pragma unroll
    for (int i = 0; i < 9; ++i) a.G[i] = p[i];
#pragma unroll
    for (int i = 0; i < 3; ++i) a.d[i] = p[9 + i];
}
__device__ __forceinline__ void aff_store(float* p, const Aff& a) {
#pragma unroll
    for (int i = 0; i < 9; ++i) p[i] = a.G[i];
#pragma unroll
    for (int i = 0; i < 3; ++i) p[9 + i] = a.d[i];
}

// ---------------- kernel 1: spherical -> local point ----------------
__global__ void k_point(const float* __restrict__ inner, float* __restrict__ pts) {
    int i = blockIdx.x * blockDim.x + threadIdx.x;
    if (i >= N_ATOMS) return;
    float len = inner[3 * i + 0];
    float ang = inner[3 * i + 1];
    float dhd = inner[3 * i + 2];   // dhd_v permutation is the identity
    float sa, ca, sd, cd;
    __sincosf(ang, &sa, &ca);
    __sincosf(dhd, &sd, &cd);
    float rc = len * ca, rs = len * sa;
    pts[3 * i + 0] = rc;
    pts[3 * i + 1] = cd * rs;
    pts[3 * i + 2] = sd * rs;
}

// ---------------- kernel 2: canonical sub-chains (depth 100, 30000-wide) ----
// INIT_MATRIX triple has frame_rows == I and origin 0, so every sub-chain is
// computed in the canonical frame; its end affine (G,d) stitches it exactly.
__global__ void k_subchain(float* __restrict__ pts /* in: points, out: canonical (in-place) */,
                           float* __restrict__ segT) {
    int s = blockIdx.x * blockDim.x + threadIdx.x;
    if (s >= NSEG) return;
    F3 A = mkf3(-0.70710678118654752f, 1.22474487139158905f, 0.f);  // INIT row 0
    F3 B = mkf3(-1.41421356237309515f, 0.f, 0.f);                   // INIT row 1
    F3 C = mkf3(0.f, 0.f, 0.f);                                     // INIT row 2
    float* q = pts + (size_t)s * SUB * 3;
    for (int k = 0; k < SUB; ++k) {
        __builtin_prefetch(q + 3 * (k + 1), 0, 0);   // global_prefetch_b8: next step's point
        float px = q[3 * k + 0], py = q[3 * k + 1], pz = q[3 * k + 2];
        F3 r0, r1, r2;
        frame_rows(A, B, C, r0, r1, r2);
        F3 D = add3(add3(add3(scl3(r0, px), scl3(r1, py)), scl3(r2, pz)), C);
        q[3 * k + 0] = D.x; q[3 * k + 1] = D.y; q[3 * k + 2] = D.z;
        A = B; B = C; C = D;
    }
    F3 r0, r1, r2;
    frame_rows(A, B, C, r0, r1, r2);
    float* t = segT + (size_t)s * 12;
    t[0] = r0.x; t[1]  = r0.y; t[2]  = r0.z;
    t[3] = r1.x; t[4]  = r1.y; t[5]  = r1.z;
    t[6] = r2.x; t[7]  = r2.y; t[8]  = r2.z;
    t[9] = C.x;  t[10] = C.y;  t[11] = C.z;
}

// ---------------- kernel 3: affine scan over 30000 segments ----------------
// chunked fold (30/thread) + Hillis-Steele inclusive scan in LDS (48 KB).
__global__ void __launch_bounds__(SCAN_T) k_scan(const float* __restrict__ segT,
                                                 const float* __restrict__ mainchain,
                                                 float* __restrict__ compT) {
    __shared__ float lds[SCAN_T * 12];
    int t = threadIdx.x;
    int s0 = t * CHUNK;

    Aff acc = aff_identity();
    for (int k = 0; k < CHUNK; ++k) {
        int s = s0 + k;
        if (s < NSEG) {
            Aff g; aff_load(g, segT + (size_t)s * 12);
            acc = aff_combine(acc, g);
        }
    }
    aff_store(&lds[t * 12], acc);
    __syncthreads();

    for (int st = 1; st < SCAN_T; st <<= 1) {
        Aff prev;
        bool has = (t >= st);
        if (has) aff_load(prev, &lds[(t - st) * 12]);
        __syncthreads();
        if (has) {
            acc = aff_combine(prev, acc);
            aff_store(&lds[t * 12], acc);
        }
        __syncthreads();
    }

    Aff pre;
    if (t == 0) pre = aff_identity();
    else        aff_load(pre, &lds[(t - 1) * 12]);

    // initial state from mainchain_coord_tensor's first 3 rows
    F3 A = mkf3(mainchain[0], mainchain[1], mainchain[2]);
    F3 B = mkf3(mainchain[3], mainchain[4], mainchain[5]);
    F3 C = mkf3(mainchain[6], mainchain[7], mainchain[8]);
    F3 r0, r1, r2;
    frame_rows(A, B, C, r0, r1, r2);
    Aff init;
    init.G[0] = r0.x; init.G[1] = r0.y; init.G[2] = r0.z;
    init.G[3] = r1.x; init.G[4] = r1.y; init.G[5] = r1.z;
    init.G[6] = r2.x; init.G[7] = r2.y; init.G[8] = r2.z;
    init.d[0] = C.x;  init.d[1] = C.y;  init.d[2] = C.z;

    Aff st = aff_combine(init, pre);   // state entering first segment of this chunk
    for (int k = 0; k < CHUNK; ++k) {
        int s = s0 + k;
        if (s < NSEG) {
            aff_store(compT + (size_t)s * 12, st);
            Aff g; aff_load(g, segT + (size_t)s * 12);
            st = aff_combine(st, g);
        }
    }
}

// ---------------- kernel 4: apply composed affines via WMMA f32 16x16x4 -----
// D = A x B : A(16x4) = affine packed column-wise, homogeneous t in K=3;
//             B(4x16) = 16 points as columns, 1.0 in K=3.
// D rows 0..2 (lanes 0..15 of d[0..2]) = x,y,z of the 16 transformed points.
// Branchless operand build: every lane does one unconditional 2-float load
// (hi lanes read z + next-x; the 1-float overread past the last point lands in
// segT within the same workspace), then one cndmask for the homogeneous 1.0.
__global__ void k_apply(const float* __restrict__ canon,
                        const float* __restrict__ compT,
                        const float* __restrict__ mainchain,
                        float* __restrict__ out) {
    int tid  = threadIdx.x;
    if (blockIdx.x == 0 && tid < 9) out[tid] = mainchain[tid];   // frame0 rows

    int wave = blockIdx.x * (blockDim.x >> 5) + (tid >> 5);      // one segment per wave
    int lane = tid & 31;
    if (wave >= NSEG) return;

    const float* T = compT + (size_t)wave * 12;
    const int  m  = lane & 15;
    const bool hi = lane >= 16;

    v2f a; a.x = 0.f; a.y = 0.f;
    if (m < 3) {
        int off = hi ? 6 : 0;       // lo: K=0/K=1 -> F rows 0,1 ; hi: K=2/K=3 -> F row 2, c
        a.x = T[off + m];
        a.y = T[off + 3 + m];
    }

    const int base = wave * SUB;
    const v8f cz = {0.f, 0.f, 0.f, 0.f, 0.f, 0.f, 0.f, 0.f};
    const size_t esel = hi ? 2u : 0u;

    // 6 full 16-point tiles: no clamp, unconditional lane<16 store
#pragma unroll
    for (int tile = 0; tile < 6; ++tile) {
        int pi = base + tile * 16 + m;
        const float* s = canon + 3 * (size_t)pi + esel;
        float x0 = s[0];
        float y0 = s[1];                   // unconditional; hi lanes discard via cndmask
        v2f b; b.x = x0; b.y = hi ? 1.0f : y0;
        v8f d = __builtin_amdgcn_wmma_f32_16x16x4_f32(
            false, a, false, b, (short)0, cz, false, false);
        if (!hi) {
            size_t o = (size_t)(3 + pi) * 3;
            out[o + 0] = d[0];
            out[o + 1] = d[1];
            out[o + 2] = d[2];
        }
    }

    // tail tile: points 96..99 (clamped), guarded store
    {
        int pi  = base + 96 + m;
        int pic = (m < 4) ? pi : (base + SUB - 1);
        const float* s = canon + 3 * (size_t)pic + esel;
        float x0 = s[0];
        float y0 = s[1];
        v2f b; b.x = x0; b.y = hi ? 1.0f : y0;
        v8f d = __builtin_amdgcn_wmma_f32_16x16x4_f32(
            false, a, false, b, (short)0, cz, false, false);
        if (!hi && m < 4) {
            size_t o = (size_t)(3 + pi) * 3;
            out[o + 0] = d[0];
            out[o + 1] = d[1];
            out[o + 2] = d[2];
        }
    }
}

// ---------------- host side ----------------
extern "C" void kernel_launch(void* const* d_in, const int* in_sizes, int n_in,
                              void* d_out, int out_size, void* d_ws, size_t ws_size,
                              hipStream_t stream) {
    const float* inner     = (const float*)d_in[0];   // (3M, 3)
    const float* mainchain = (const float*)d_in[2];   // (3, 3)
    float* out = (float*)d_out;                       // (3M+3, 3)
    float* ws  = (float*)d_ws;

    float* pts   = ws;                                   // 9,000,000 f32 (points -> canonical, in place)
    float* segT  = ws + (size_t)N_ATOMS * 3;             // 30000 * 12 f32
    float* compT = segT + (size_t)NSEG * 12;             // 30000 * 12 f32

    k_point   <<<(N_ATOMS + 255) / 256, 256, 0, stream>>>(inner, pts);
    k_subchain<<<(NSEG    + 255) / 256, 256, 0, stream>>>(pts, segT);
    k_scan    <<<1, SCAN_T, 0, stream>>>(segT, mainchain, compT);
    k_apply   <<<(NSEG + 7) / 8, 256, 0, stream>>>(pts, compT, mainchain, out);
}